// monet_segmentation_19524921328231
// MI455X (gfx1250) — compile-verified
//
#include <hip/hip_runtime.h>
#include <hip/hip_bf16.h>

// ---------------------------------------------------------------------------
// MoNet (GMMConv) icosphere U-Net, fp32, WMMA F32_16X16X4_F32 GEMMs (gfx1250)
// ---------------------------------------------------------------------------

typedef float v2f __attribute__((ext_vector_type(2)));
typedef float v8f __attribute__((ext_vector_type(8)));

#define GEMM_KC 32   // K chunk (8 wmma steps of K=4)

static __device__ __forceinline__ int imin(int a, int b) { return a < b ? a : b; }

// C[N,Co] = A[N,Ci] @ B[Ci,Co] (+bias). Block = 128 threads = 4 waves.
// Each wave computes one 16x16 tile. In the 16x4 f32 A layout, lane l needs
// A[row0+l][k+2*half .. +1] -- a contiguous 8B-aligned float2 from the lane's
// own row -- so fragments load straight from global (no LDS, no barriers).
// Addressing uses uniform SGPR bases advanced per chunk + loop-invariant
// per-lane 32-bit offsets (GVS mode): zero per-chunk VALU address math.
// Per chunk: 8 A + 8 B fragment loads in flight, then 8 WMMAs back-to-back.
// Out-of-range rows/cols are address-clamped (garbage only reaches unstored
// output rows/cols); the single tail chunk zero-pads K on the A side.
// EXEC stays all-ones through every WMMA.
__global__ __launch_bounds__(128)
void monet_gemm_wmma_f32(const float* __restrict__ A, const float* __restrict__ B,
                         const float* __restrict__ bias, float* __restrict__ C,
                         int N, int Ci, int Co)
{
    const int wave = threadIdx.x >> 5;
    const int lane = threadIdx.x & 31;
    const int half = lane >> 4;          // 0: lanes 0-15, 1: lanes 16-31
    const int l    = lane & 15;

    const int row0 = blockIdx.y * 16;
    const int col0 = (blockIdx.x * 4 + wave) * 16;
    const int col  = col0 + l;
    const int colc = imin(col, Co - 1);      // clamped (garbage cols never stored)
    const int arow = imin(row0 + l, N - 1);  // clamped (garbage rows never stored)

    // loop-invariant per-lane element offsets (32-bit; buffers are < 4GB)
    const int aoff = arow * Ci + 2 * half;   // lane's A row, K=2*half
    int boff[16];
    #pragma unroll
    for (int kk = 0; kk < 8; ++kk) {
        boff[2 * kk + 0] = (4 * kk + 0 + 2 * half) * Co + colc;
        boff[2 * kk + 1] = (4 * kk + 1 + 2 * half) * Co + colc;
    }

    // uniform (SGPR) base pointers, advanced by scalar adds per chunk
    const float* ab = A;
    const float* bb = B;

    v8f acc = {};
    int k0 = 0;

    // ---- main loop: full K chunks, unconditional saddr+voffset loads ----
    for (; k0 + GEMM_KC <= Ci; k0 += GEMM_KC) {
        v2f a[8], b[8];
        #pragma unroll
        for (int kk = 0; kk < 8; ++kk) {
            a[kk] = *(const v2f*)&ab[aoff + 4 * kk];     // global_load_b64
            b[kk].x = bb[boff[2 * kk + 0]];
            b[kk].y = bb[boff[2 * kk + 1]];
        }
        ab += GEMM_KC;                                   // uniform scalar adds
        bb += (size_t)GEMM_KC * Co;
        __builtin_prefetch(&bb[boff[0]], 0, 1);          // next B chunk

        #pragma unroll
        for (int kk = 0; kk < 8; ++kk)
            acc = __builtin_amdgcn_wmma_f32_16x16x4_f32(
                    false, a[kk], false, b[kk], (short)0, acc, false, false);
    }

    // ---- tail chunk (Ci % GEMM_KC): clamped addresses, K zero-pad on A ----
    if (k0 < Ci) {
        v2f a[8], b[8];
        #pragma unroll
        for (int kk = 0; kk < 8; ++kk) {
            int ka = k0 + 4 * kk + 2 * half;
            float ax = A[(size_t)arow * Ci + imin(ka, Ci - 1)];
            float ay = A[(size_t)arow * Ci + imin(ka + 1, Ci - 1)];
            a[kk].x = (ka < Ci) ? ax : 0.0f;             // zero A side for k >= Ci
            a[kk].y = (ka + 1 < Ci) ? ay : 0.0f;
            b[kk].x = B[(size_t)imin(ka, Ci - 1) * Co + colc];
            b[kk].y = B[(size_t)imin(ka + 1, Ci - 1) * Co + colc];
        }
        #pragma unroll
        for (int kk = 0; kk < 8; ++kk)
            acc = __builtin_amdgcn_wmma_f32_16x16x4_f32(
                    false, a[kk], false, b[kk], (short)0, acc, false, false);
    }

    // C 16x16 layout: VGPR v -> row v (lanes 0-15) / row v+8 (lanes 16-31)
    if (col < Co) {
        float bv = bias ? bias[col] : 0.0f;
        #pragma unroll
        for (int v = 0; v < 8; ++v) {
            int row = row0 + v + 8 * half;
            if (row < N) C[(size_t)row * Co + col] = acc[v] + bv;
        }
    }
}

__global__ void monet_fill_zero(float* __restrict__ p, int n)
{
    int i = blockIdx.x * blockDim.x + threadIdx.x;
    if (i < n) p[i] = 0.0f;
}

// One block (64 threads) per edge; threads stride output channels.
__global__ __launch_bounds__(64)
void monet_edge_msg(const float* __restrict__ xt,     // [N, 3*Cout]
                    const int* __restrict__ src, const int* __restrict__ dst,
                    const float* __restrict__ pseudo,  // [E,2]
                    const float* __restrict__ mu,      // [3,2]
                    const float* __restrict__ sigma,   // [3,2]
                    float* __restrict__ agg,           // [N, Cout]
                    float* __restrict__ deg,           // [N]
                    int Cout)
{
    const int e = blockIdx.x;
    const float p0 = pseudo[2 * e + 0];
    const float p1 = pseudo[2 * e + 1];
    const int s = src[e];
    const int d = dst[e];

    float w[3];
    #pragma unroll
    for (int k = 0; k < 3; ++k) {
        float d0 = p0 - mu[2 * k + 0];
        float d1 = p1 - mu[2 * k + 1];
        float s0 = sigma[2 * k + 0];
        float s1 = sigma[2 * k + 1];
        w[k] = __expf(-0.5f * (d0 * d0 / (1e-15f + s0 * s0) +
                               d1 * d1 / (1e-15f + s1 * s1)));
    }

    const float* xs = xt + (size_t)s * (3 * Cout);
    float* ag = agg + (size_t)d * Cout;
    for (int m = threadIdx.x; m < Cout; m += 64) {
        float acc = xs[m] * w[0] + xs[Cout + m] * w[1] + xs[2 * Cout + m] * w[2];
        atomicAdd(&ag[m], acc);
    }
    if (threadIdx.x == 0) atomicAdd(&deg[d], 1.0f);
}

// One block (64 threads) per node: out = relu?( agg/max(deg,1) + r )
__global__ __launch_bounds__(64)
void monet_combine(const float* __restrict__ agg, const float* __restrict__ deg,
                   const float* __restrict__ r, float* __restrict__ out,
                   int Cout, int relu)
{
    const int n = blockIdx.x;
    const float inv = 1.0f / fmaxf(deg[n], 1.0f);
    for (int m = threadIdx.x; m < Cout; m += 64) {
        size_t idx = (size_t)n * Cout + m;
        float v = agg[idx] * inv + r[idx];
        out[idx] = relu ? fmaxf(v, 0.0f) : v;
    }
}

// out[hi, C+Cs] = concat(hex_up(h[lo,C], up), skip[hi,Cs])
__global__ void monet_upconcat(const float* __restrict__ h, const int* __restrict__ up,
                               const float* __restrict__ skip, float* __restrict__ out,
                               int lo, int hi, int C, int Cs)
{
    int tot = hi * (C + Cs);
    int tid = blockIdx.x * blockDim.x + threadIdx.x;
    if (tid >= tot) return;
    int n = tid / (C + Cs);
    int c = tid - n * (C + Cs);
    float v;
    if (c < C) {
        if (n < lo) {
            v = h[(size_t)n * C + c];
        } else {
            int i = n - lo;
            int p0 = up[2 * i + 0], p1 = up[2 * i + 1];
            v = 0.5f * (h[(size_t)p0 * C + c] + h[(size_t)p1 * C + c]);
        }
    } else {
        v = skip[(size_t)n * Cs + (c - C)];
    }
    out[tid] = v;
}

__global__ void monet_softmax_rows(float* __restrict__ x, int N, int C)
{
    int n = blockIdx.x * blockDim.x + threadIdx.x;
    if (n >= N) return;
    float* row = x + (size_t)n * C;
    float mx = -3.402823466e38f;
    for (int c = 0; c < C; ++c) mx = fmaxf(mx, row[c]);
    float s = 0.0f;
    for (int c = 0; c < C; ++c) { float e = __expf(row[c] - mx); row[c] = e; s += e; }
    float inv = 1.0f / s;
    for (int c = 0; c < C; ++c) row[c] *= inv;
}

// ---------------------------------------------------------------------------
// host orchestration
// ---------------------------------------------------------------------------
static inline int ceil_div(int a, int b) { return (a + b - 1) / b; }

static void run_gmm(const float* x, int N, int Cin, int Cout,
                    const float* g, const float* mu, const float* sigma,
                    const float* root, const float* bias,
                    const int* edge, const float* pseudo, int E,
                    float* out, int relu,
                    float* xt, float* rr, float* agg, float* deg,
                    hipStream_t stream)
{
    const int Co3 = 3 * Cout;
    // xt = x @ g   [N, 3*Cout]
    {
        dim3 grid(ceil_div(Co3, 64), ceil_div(N, 16));
        monet_gemm_wmma_f32<<<grid, 128, 0, stream>>>(x, g, nullptr, xt, N, Cin, Co3);
    }
    // rr = x @ root + bias   [N, Cout]
    {
        dim3 grid(ceil_div(Cout, 64), ceil_div(N, 16));
        monet_gemm_wmma_f32<<<grid, 128, 0, stream>>>(x, root, bias, rr, N, Cin, Cout);
    }
    // zero agg, deg
    monet_fill_zero<<<ceil_div(N * Cout, 256), 256, 0, stream>>>(agg, N * Cout);
    monet_fill_zero<<<ceil_div(N, 256), 256, 0, stream>>>(deg, N);
    // edge messages -> agg, deg (block per edge)
    monet_edge_msg<<<E, 64, 0, stream>>>(xt, edge, edge + E, pseudo, mu, sigma,
                                         agg, deg, Cout);
    // combine (block per node)
    monet_combine<<<N, 64, 0, stream>>>(agg, deg, rr, out, Cout, relu);
}

extern "C" void kernel_launch(void* const* d_in, const int* in_sizes, int n_in,
                              void* d_out, int out_size, void* d_ws, size_t ws_size,
                              hipStream_t stream)
{
    (void)in_sizes; (void)n_in; (void)out_size; (void)ws_size;

    // input layout: x, 11x(g,mu,sigma,root,bias), edges[0..5], pseudos[0..5], ups[0..4]
    const float* x0 = (const float*)d_in[0];                       // [40962,4]
    const float* G[11];  const float* MU[11]; const float* SG[11];
    const float* RT[11]; const float* BS[11];
    for (int i = 0; i < 11; ++i) {
        G[i]  = (const float*)d_in[1 + 5 * i];
        MU[i] = (const float*)d_in[2 + 5 * i];
        SG[i] = (const float*)d_in[3 + 5 * i];
        RT[i] = (const float*)d_in[4 + 5 * i];
        BS[i] = (const float*)d_in[5 + 5 * i];
    }
    const int*   ED[6]; const float* PS[6];
    for (int l = 0; l < 6; ++l) {
        ED[l] = (const int*)d_in[56 + l];
        PS[l] = (const float*)d_in[62 + l];
    }
    const int* UP[5];
    for (int j = 0; j < 5; ++j) UP[j] = (const int*)d_in[68 + j];

    static const int Elev[6] = {240, 960, 3840, 15360, 61440, 245760};

    // bump allocator over workspace
    float* ws = (float*)d_ws;
    size_t off = 0;
    auto alloc = [&](size_t n) { float* p = ws + off; off += (n + 63) & ~(size_t)63; return p; };

    float* h6   = alloc((size_t)40962 * 64);    // conv1 out (x1 = prefix 10242 rows)
    float* h5   = alloc((size_t)10242 * 128);   // conv2 out (x2 = prefix 2562)
    float* h4   = alloc((size_t)2562  * 256);   // conv3 out (x3 = prefix 642)
    float* h3   = alloc((size_t)642   * 512);   // conv4 out (x4 = prefix 162)
    float* h2   = alloc((size_t)162   * 1024);  // conv5 out (x5 = prefix 42)
    float* h1   = alloc((size_t)42    * 1024);  // conv6 out
    float* in7  = alloc((size_t)162   * 1536);
    float* o7   = alloc((size_t)162   * 512);
    float* in8  = alloc((size_t)642   * 768);
    float* o8   = alloc((size_t)642   * 256);
    float* in9  = alloc((size_t)2562  * 384);
    float* o9   = alloc((size_t)2562  * 128);
    float* in10 = alloc((size_t)10242 * 192);
    float* o10  = alloc((size_t)10242 * 64);
    float* in11 = alloc((size_t)40962 * 68);
    float* xt   = alloc((size_t)40962 * 192);   // max over all convs
    float* rr   = alloc((size_t)40962 * 64);
    float* agg  = alloc((size_t)40962 * 64);
    float* deg  = alloc((size_t)40962);

    // ---- encoder ----
    run_gmm(x0, 40962,    4,   64, G[0], MU[0], SG[0], RT[0], BS[0], ED[5], PS[5], Elev[5], h6, 1, xt, rr, agg, deg, stream);
    run_gmm(h6, 10242,   64,  128, G[1], MU[1], SG[1], RT[1], BS[1], ED[4], PS[4], Elev[4], h5, 1, xt, rr, agg, deg, stream);
    run_gmm(h5,  2562,  128,  256, G[2], MU[2], SG[2], RT[2], BS[2], ED[3], PS[3], Elev[3], h4, 1, xt, rr, agg, deg, stream);
    run_gmm(h4,   642,  256,  512, G[3], MU[3], SG[3], RT[3], BS[3], ED[2], PS[2], Elev[2], h3, 1, xt, rr, agg, deg, stream);
    run_gmm(h3,   162,  512, 1024, G[4], MU[4], SG[4], RT[4], BS[4], ED[1], PS[1], Elev[1], h2, 1, xt, rr, agg, deg, stream);
    run_gmm(h2,    42, 1024, 1024, G[5], MU[5], SG[5], RT[5], BS[5], ED[0], PS[0], Elev[0], h1, 1, xt, rr, agg, deg, stream);

    // ---- decoder ----
    monet_upconcat<<<ceil_div(162 * 1536, 256), 256, 0, stream>>>(h1, UP[0], h3, in7, 42, 162, 1024, 512);
    run_gmm(in7,   162, 1536,  512, G[6],  MU[6],  SG[6],  RT[6],  BS[6],  ED[1], PS[1], Elev[1], o7,  1, xt, rr, agg, deg, stream);

    monet_upconcat<<<ceil_div(642 * 768, 256), 256, 0, stream>>>(o7, UP[1], h4, in8, 162, 642, 512, 256);
    run_gmm(in8,   642,  768,  256, G[7],  MU[7],  SG[7],  RT[7],  BS[7],  ED[2], PS[2], Elev[2], o8,  1, xt, rr, agg, deg, stream);

    monet_upconcat<<<ceil_div(2562 * 384, 256), 256, 0, stream>>>(o8, UP[2], h5, in9, 642, 2562, 256, 128);
    run_gmm(in9,  2562,  384,  128, G[8],  MU[8],  SG[8],  RT[8],  BS[8],  ED[3], PS[3], Elev[3], o9,  1, xt, rr, agg, deg, stream);

    monet_upconcat<<<ceil_div(10242 * 192, 256), 256, 0, stream>>>(o9, UP[3], h6, in10, 2562, 10242, 128, 64);
    run_gmm(in10, 10242, 192,   64, G[9],  MU[9],  SG[9],  RT[9],  BS[9],  ED[4], PS[4], Elev[4], o10, 1, xt, rr, agg, deg, stream);

    monet_upconcat<<<ceil_div(40962 * 68, 256), 256, 0, stream>>>(o10, UP[4], x0, in11, 10242, 40962, 64, 4);
    run_gmm(in11, 40962,  68,   37, G[10], MU[10], SG[10], RT[10], BS[10], ED[5], PS[5], Elev[5],
            (float*)d_out, 0, xt, rr, agg, deg, stream);

    // ---- softmax over channels ----
    monet_softmax_rows<<<ceil_div(40962, 128), 128, 0, stream>>>((float*)d_out, 40962, 37);
}